// StochasticTwoLayerGCN_65111704207519
// MI455X (gfx1250) — compile-verified
//
#include <hip/hip_runtime.h>

// Problem constants (fixed by the reference)
#define N0_SRC 819200
#define N0_DST 81920
#define E0     819200
#define N1_SRC 81920
#define N1_DST 8192
#define E1     81920
#define IN_F   128
#define HID_F  128
#define OUT_F  64

typedef __attribute__((ext_vector_type(2))) float v2f;
typedef __attribute__((ext_vector_type(8))) float v8f;

// ---------------------------------------------------------------- utilities
__global__ void zero_f32(float* __restrict__ p, long n) {
    long i = (long)blockIdx.x * blockDim.x + threadIdx.x;
    long stride = (long)gridDim.x * blockDim.x;
    for (; i < n; i += stride) p[i] = 0.0f;
}

__global__ void count_deg(const int* __restrict__ idx, int E, float* __restrict__ deg) {
    int i = blockIdx.x * blockDim.x + threadIdx.x;
    if (i < E) atomicAdd(deg + idx[i], 1.0f);
}

__global__ void finalize_rsqrt(float* __restrict__ p, int n) {
    int i = blockIdx.x * blockDim.x + threadIdx.x;
    if (i < n) {
        float d = p[i];
        d = d < 1.0f ? 1.0f : d;     // clip(deg, 1)
        p[i] = rsqrtf(d);
    }
}

// ------------------------------------------------- edge gather + scatter-add
// One thread handles 4 features of one edge: 32 consecutive threads cover one
// edge's full 128-float row (coalesced float4 gather, L2-resident atomics).
__global__ void scatter_edges(const float* __restrict__ feat,
                              const float* __restrict__ rs_src,
                              const int* __restrict__ src,
                              const int* __restrict__ dst,
                              int E, float* __restrict__ agg) {
    long tid = (long)blockIdx.x * blockDim.x + threadIdx.x;
    long total = (long)E * 32;
    if (tid >= total) return;
    int e = (int)(tid >> 5);
    int c = ((int)tid & 31) << 2;
    int s = src[e];
    int d = dst[e];
    float r = rs_src[s];             // src-side norm folded into the gather
    const float4 v = *(const float4*)(feat + (long)s * 128 + c);
    float* out = agg + (long)d * 128 + c;
    atomicAdd(out + 0, v.x * r);
    atomicAdd(out + 1, v.y * r);
    atomicAdd(out + 2, v.z * r);
    atomicAdd(out + 3, v.w * r);
}

// ------------------------------------------ fused rownorm + GEMM + bias+ReLU
// C[M,N] = relu( (A * rs[:,None]) @ W + bias ), fp32 via V_WMMA_F32_16X16X4_F32.
// 8 waves per block; each wave computes one 16x16 output tile.
// A-matrix 16x4 layout : lane L -> row L%16, K pair 2*(L/16)
// B-matrix 4x16 layout : lane L -> col L%16, K pair 2*(L/16)
// C/D 16x16 layout     : vgpr v, lane L -> (m,n) = (v + 8*(L/16), L%16)
__global__ __launch_bounds__(256)
void gemm_rownorm_wmma(const float* __restrict__ A,
                       const float* __restrict__ rs,
                       const float* __restrict__ W,
                       const float* __restrict__ bias,
                       float* __restrict__ C,
                       int N, int K, int tilesN) {
    int wid  = threadIdx.x >> 5;
    int tile = blockIdx.x * 8 + wid;
    int tm   = tile / tilesN;
    int tn   = tile - tm * tilesN;
    int lane = threadIdx.x & 31;
    int half = lane >> 4;            // 0: K={0,1}, 1: K={2,3} within a k-step
    int l16  = lane & 15;

    long rowA = (long)tm * 16 + l16;
    int  colB = tn * 16 + l16;
    float r = rs[rowA];              // dst-side norm folded into A
    const float* Arow = A + rowA * (long)K;

    v8f acc = {};
    for (int k = 0; k < K; k += 4) {
        int ka = k + half * 2;
        float2 av = *(const float2*)(Arow + ka);   // 8B-aligned contiguous pair
        v2f a; a.x = av.x * r; a.y = av.y * r;
        v2f b;
        b.x = W[(long)ka * N + colB];
        b.y = W[(long)(ka + 1) * N + colB];
        acc = __builtin_amdgcn_wmma_f32_16x16x4_f32(
            /*neg_a=*/false, a, /*neg_b=*/false, b,
            /*c_mod=*/(short)0, acc, /*reuse_a=*/false, /*reuse_b=*/false);
    }

    float bv = bias[colB];
#pragma unroll
    for (int v = 0; v < 8; ++v) {
        long cm = (long)tm * 16 + v + 8 * half;
        float val = acc[v] + bv;
        C[cm * (long)N + colB] = val > 0.0f ? val : 0.0f;
    }
}

// ---------------------------------------------------------------- launcher
extern "C" void kernel_launch(void* const* d_in, const int* in_sizes, int n_in,
                              void* d_out, int out_size, void* d_ws, size_t ws_size,
                              hipStream_t stream) {
    const float* x   = (const float*)d_in[0];
    const float* W0  = (const float*)d_in[1];
    const float* b0  = (const float*)d_in[2];
    const float* W1  = (const float*)d_in[3];
    const float* b1  = (const float*)d_in[4];
    const int*   e0s = (const int*)d_in[5];
    const int*   e0d = (const int*)d_in[6];
    const int*   e1s = (const int*)d_in[7];
    const int*   e1d = (const int*)d_in[8];

    // Workspace layout (floats)
    float* ws   = (float*)d_ws;
    float* rs0s = ws;                                   // N0_SRC
    float* rs0d = rs0s + N0_SRC;                        // N0_DST
    float* rs1s = rs0d + N0_DST;                        // N1_SRC
    float* rs1d = rs1s + N1_SRC;                        // N1_DST
    float* agg0 = rs1d + N1_DST;                        // N0_DST * HID_F
    float* agg1 = agg0 + (size_t)N0_DST * HID_F;        // N1_DST * HID_F
    float* h1   = agg1 + (size_t)N1_DST * HID_F;        // N0_DST * HID_F

    const int degN = N0_SRC + N0_DST + N1_SRC + N1_DST;
    const long zeroN = (long)degN + (long)N0_DST * HID_F + (long)N1_DST * HID_F;

    // 1) zero degree accumulators + aggregation buffers (replay-safe)
    zero_f32<<<4096, 256, 0, stream>>>(ws, zeroN);

    // 2) degrees
    count_deg<<<(E0 + 255) / 256, 256, 0, stream>>>(e0s, E0, rs0s);
    count_deg<<<(E0 + 255) / 256, 256, 0, stream>>>(e0d, E0, rs0d);
    count_deg<<<(E1 + 255) / 256, 256, 0, stream>>>(e1s, E1, rs1s);
    count_deg<<<(E1 + 255) / 256, 256, 0, stream>>>(e1d, E1, rs1d);
    finalize_rsqrt<<<(degN + 255) / 256, 256, 0, stream>>>(ws, degN);

    // 3) layer 0: gather x * rs0s along e0, scatter-add into agg0
    {
        long tot = (long)E0 * 32;
        scatter_edges<<<(int)((tot + 255) / 256), 256, 0, stream>>>(x, rs0s, e0s, e0d, E0, agg0);
    }

    // 4) h1 = relu((agg0 * rs0d) @ W0 + b0)   [81920 x 128] via f32 WMMA
    {
        int tiles = (N0_DST / 16) * (HID_F / 16);       // 40960
        gemm_rownorm_wmma<<<tiles / 8, 256, 0, stream>>>(agg0, rs0d, W0, b0, h1,
                                                         HID_F, IN_F, HID_F / 16);
    }

    // 5) layer 1: gather h1 * rs1s along e1, scatter-add into agg1
    {
        long tot = (long)E1 * 32;
        scatter_edges<<<(int)((tot + 255) / 256), 256, 0, stream>>>(h1, rs1s, e1s, e1d, E1, agg1);
    }

    // 6) out = relu((agg1 * rs1d) @ W1 + b1)  [8192 x 64] via f32 WMMA
    {
        int tiles = (N1_DST / 16) * (OUT_F / 16);       // 2048
        gemm_rownorm_wmma<<<tiles / 8, 256, 0, stream>>>(agg1, rs1d, W1, b1, (float*)d_out,
                                                         OUT_F, HID_F, OUT_F / 16);
    }
}